// DotProductNonLocalMeans_3092376453294
// MI455X (gfx1250) — compile-verified
//
#include <hip/hip_runtime.h>
#include <hip/hip_bf16.h>

// ---------------------------------------------------------------------------
// out[b] = (W @ (X_b X_b^T / N)) @ X_b,  X_b = x[b] viewed as [64, 16384] fp32
// Bandwidth-bound (~128MB HBM floor -> ~5.5us @ 23.3TB/s).
//  - fp32 WMMA (v_wmma_f32_16x16x4_f32) = exact reference precision
//  - global_load_async_to_lds_b128 streams tiles memory->LDS (ASYNCcnt path,
//    no VGPR staging / loadcnt stalls)
//  - non-temporal output stores keep x resident in the 192MB L2 for pass 3
// ---------------------------------------------------------------------------

typedef __attribute__((ext_vector_type(2))) float v2f;
typedef __attribute__((ext_vector_type(8))) float v8f;

#define B_DIM   16
#define C_DIM   64
#define N_DIM   16384
#define NSPLIT  32           // N-splits per batch for gram partials
#define CHUNK   128          // columns staged in LDS per iteration
#define CPB     512          // columns per gram block (4 chunks)
#define PITCH   132          // LDS row pitch in floats (128 + 4 pad -> conflict-free)
#define MP      68           // LDS pitch for 64-wide M matrix (64 + 4 pad)
#define NT      128          // columns per apply block

__device__ __forceinline__ unsigned lds_off(const void* p) {
    // Low 32 bits of a flat LDS address are the LDS byte offset.
    return (unsigned)(unsigned long long)p;
}

__device__ __forceinline__ void async_copy_b128(const float* gsrc, float* ldst) {
    asm volatile("global_load_async_to_lds_b128 %0, %1, off"
                 :: "v"(lds_off(ldst)), "v"(gsrc)
                 : "memory");
}

__device__ __forceinline__ void wait_async0() {
    asm volatile("s_wait_asynccnt 0x0" ::: "memory");
}

// ------------------------- Kernel 1: partial grams --------------------------
// grid = B*NSPLIT blocks, 128 threads (4 waves). Each block: 512 columns.
// Wave w computes G[16w:16w+16][0:64] partial via WMMA f32 16x16x4.
__global__ __launch_bounds__(128) void gram_partial_kernel(
    const float* __restrict__ x, float* __restrict__ gpart)
{
    __shared__ float lds[C_DIM * PITCH];

    const int b    = blockIdx.x / NSPLIT;
    const int s    = blockIdx.x % NSPLIT;
    const int tid  = threadIdx.x;
    const int lane = tid & 31;
    const int wave = tid >> 5;

    const float* xb = x + (size_t)b * C_DIM * N_DIM;

    const int row_sel = lane & 15;          // M (or B-col) index within tile
    const int ksel    = (lane >> 4) << 1;   // lanes 16-31 hold K=2,3
    const int c0      = wave * 16;

    v8f acc[4];
    const v8f vzero = {0.f, 0.f, 0.f, 0.f, 0.f, 0.f, 0.f, 0.f};
#pragma unroll
    for (int j = 0; j < 4; ++j) acc[j] = vzero;

    for (int chunk = 0; chunk < CPB / CHUNK; ++chunk) {
        const int col0 = s * CPB + chunk * CHUNK;
        // Async stream: 64 rows x 128 cols = 2048 x b128; 16 per thread.
#pragma unroll
        for (int i = 0; i < 16; ++i) {
            int e   = tid + i * 128;        // 0..2047
            int row = e >> 5;               // 32 float4 per row
            int q   = e & 31;
            async_copy_b128(xb + (size_t)row * N_DIM + col0 + q * 4,
                            &lds[row * PITCH + q * 4]);
        }
        wait_async0();
        __syncthreads();

        for (int kk = 0; kk < CHUNK; kk += 4) {
            // A tile: X rows [c0, c0+16), K slice kk..kk+3 (split across lane halves)
            v2f a = *(const v2f*)(&lds[(c0 + row_sel) * PITCH + kk + ksel]);
#pragma unroll
            for (int j = 0; j < 4; ++j) {
                // B tile = transposed X tile: B[k][d] = X[16j+d][kk+k]
                v2f bm = *(const v2f*)(&lds[(j * 16 + row_sel) * PITCH + kk + ksel]);
                acc[j] = __builtin_amdgcn_wmma_f32_16x16x4_f32(
                    false, a, false, bm, (short)0, acc[j], false, false);
            }
        }
        __syncthreads();
    }

    // gpart[(s*B + b)][row][col]
    float* gp = gpart + (size_t)(s * B_DIM + b) * C_DIM * C_DIM;
    const int mhi = (lane >> 4) << 3;       // +8 rows for upper lane half
#pragma unroll
    for (int j = 0; j < 4; ++j) {
        int col = j * 16 + row_sel;
#pragma unroll
        for (int v = 0; v < 8; ++v) {
            gp[(c0 + v + mhi) * C_DIM + col] = acc[j][v];
        }
    }
}

// --------------- Kernel 2: reduce partials, fold conv weight ----------------
// grid = B, 256 threads. M[b] = W @ (sum_s gpart[s][b]) / N   (tiny)
__global__ __launch_bounds__(256) void reduce_conv_kernel(
    const float* __restrict__ gpart, const float* __restrict__ w,
    float* __restrict__ mmat)
{
    __shared__ float gl[C_DIM * C_DIM];
    const int b   = blockIdx.x;
    const int tid = threadIdx.x;

    for (int e = tid; e < C_DIM * C_DIM; e += 256) {
        float s = 0.f;
        for (int p = 0; p < NSPLIT; ++p)
            s += gpart[(size_t)(p * B_DIM + b) * C_DIM * C_DIM + e];
        gl[e] = s * (1.0f / (float)N_DIM);
    }
    __syncthreads();

    for (int e = tid; e < C_DIM * C_DIM; e += 256) {
        int o = e >> 6, d = e & 63;
        float s = 0.f;
#pragma unroll 8
        for (int c = 0; c < C_DIM; ++c)
            s += w[o * C_DIM + c] * gl[c * C_DIM + d];
        mmat[(size_t)b * C_DIM * C_DIM + e] = s;
    }
}

// --------------------- Kernel 3: out[b] = M[b] @ X_b ------------------------
// grid = B*(N/NT), 256 threads (8 waves): 4 row strips x 2 column halves.
__global__ __launch_bounds__(256) void apply_kernel(
    const float* __restrict__ x, const float* __restrict__ mmat,
    float* __restrict__ out)
{
    __shared__ float ml[C_DIM * MP];     // M[b], padded rows
    __shared__ float xl[C_DIM * PITCH];  // X chunk [64][128], padded rows

    const int nblk = N_DIM / NT;         // 128
    const int b    = blockIdx.x / nblk;
    const int n0   = (blockIdx.x % nblk) * NT;
    const int tid  = threadIdx.x;
    const int lane = tid & 31;
    const int wave = tid >> 5;

    // Async stream X chunk: 64 x 128 = 2048 x b128; 8 per thread.
    const float* xb = x + (size_t)b * C_DIM * N_DIM;
#pragma unroll
    for (int i = 0; i < 8; ++i) {
        int e   = tid + i * 256;
        int row = e >> 5;
        int q   = e & 31;
        async_copy_b128(xb + (size_t)row * N_DIM + n0 + q * 4,
                        &xl[row * PITCH + q * 4]);
    }

    // Load M[b] (4096 floats) through registers (small, L2-resident).
    const float* mb = mmat + (size_t)b * C_DIM * C_DIM;
    for (int e = tid; e < C_DIM * C_DIM; e += 256)
        ml[(e >> 6) * MP + (e & 63)] = mb[e];

    wait_async0();
    __syncthreads();

    const int c0      = (wave >> 1) * 16;   // output-row strip
    const int nb0     = (wave & 1) * 64;    // column half
    const int row_sel = lane & 15;
    const int ksel    = (lane >> 4) << 1;

    v8f acc[4];
    const v8f vzero = {0.f, 0.f, 0.f, 0.f, 0.f, 0.f, 0.f, 0.f};
#pragma unroll
    for (int j = 0; j < 4; ++j) acc[j] = vzero;

    for (int kk = 0; kk < C_DIM; kk += 4) {
        // A tile: M rows [c0, c0+16), K = channel slice
        v2f a = *(const v2f*)(&ml[(c0 + row_sel) * MP + kk + ksel]);
#pragma unroll
        for (int j = 0; j < 4; ++j) {
            int col = nb0 + j * 16 + row_sel;
            v2f bm;
            bm.x = xl[(kk + ksel)     * PITCH + col];
            bm.y = xl[(kk + ksel + 1) * PITCH + col];
            acc[j] = __builtin_amdgcn_wmma_f32_16x16x4_f32(
                false, a, false, bm, (short)0, acc[j], false, false);
        }
    }

    // Non-temporal stores: output is write-once, keep L2 for x.
    float* ob = out + (size_t)b * C_DIM * N_DIM;
    const int mhi = (lane >> 4) << 3;
#pragma unroll
    for (int j = 0; j < 4; ++j) {
        int col = n0 + nb0 + j * 16 + row_sel;
#pragma unroll
        for (int v = 0; v < 8; ++v) {
            __builtin_nontemporal_store(acc[j][v],
                &ob[(size_t)(c0 + v + mhi) * N_DIM + col]);
        }
    }
}

// ---------------------------------------------------------------------------
extern "C" void kernel_launch(void* const* d_in, const int* in_sizes, int n_in,
                              void* d_out, int out_size, void* d_ws, size_t ws_size,
                              hipStream_t stream) {
    (void)in_sizes; (void)n_in; (void)out_size; (void)ws_size;
    const float* x = (const float*)d_in[0];   // [16,64,128,128] fp32
    const float* w = (const float*)d_in[1];   // [64,64] fp32
    float* out = (float*)d_out;               // [16,64,128,128] fp32

    float* gpart = (float*)d_ws;                                        // 8 MB
    float* mmat  = gpart + (size_t)NSPLIT * B_DIM * C_DIM * C_DIM;      // 256 KB

    gram_partial_kernel<<<B_DIM * NSPLIT, 128, 0, stream>>>(x, gpart);
    reduce_conv_kernel<<<B_DIM, 256, 0, stream>>>(gpart, w, mmat);
    apply_kernel<<<B_DIM * (N_DIM / NT), 256, 0, stream>>>(x, mmat, out);
}